// PointNetfeat_63909113364508
// MI455X (gfx1250) — compile-verified
//
#include <hip/hip_runtime.h>
#include <hip/hip_bf16.h>

typedef __attribute__((ext_vector_type(2))) float v2f;
typedef __attribute__((ext_vector_type(8))) float v8f;

// ---------------------------------------------------------------------------
// fp32 WMMA: D = A(16x4) * B(4x16) + C(16x16), wave32.
// A layout: lanes 0-15 -> M=lane, VGPR0 holds K0 (lanes0-15)/K2 (lanes16-31),
//           VGPR1 holds K1/K3.  C/D: VGPR r -> M=r (lanes0-15), M=r+8 (16-31).
// ---------------------------------------------------------------------------
__device__ __forceinline__ v8f wmma4(v2f a, v2f b, v8f c) {
  return __builtin_amdgcn_wmma_f32_16x16x4_f32(
      /*neg_a=*/false, a, /*neg_b=*/false, b,
      /*c_mod=*/(short)0, c, /*reuse_a=*/false, /*reuse_b=*/false);
}

// Order-preserving float <-> uint mapping for atomicMax on floats.
__device__ __forceinline__ unsigned fenc(float x) {
  unsigned u = __float_as_uint(x);
  return (u & 0x80000000u) ? ~u : (u | 0x80000000u);
}
__device__ __forceinline__ float fdec(unsigned e) {
  return __uint_as_float((e & 0x80000000u) ? (e & 0x7FFFFFFFu) : ~e);
}

// ---------------------------------------------------------------------------
// Utility kernels
// ---------------------------------------------------------------------------
__global__ void zero_f32_kernel(float* __restrict__ p, int n) {
  int i = blockIdx.x * blockDim.x + threadIdx.x;
  if (i < n) p[i] = 0.0f;
}
__global__ void zero_u32_kernel(unsigned* __restrict__ p, int n) {
  int i = blockIdx.x * blockDim.x + threadIdx.x;
  if (i < n) p[i] = 0u;   // fdec(0) is below any real encoded value
}
__global__ void decode_kernel(const unsigned* __restrict__ e, float* __restrict__ g, int n) {
  int i = blockIdx.x * blockDim.x + threadIdx.x;
  if (i < n) g[i] = fdec(e[i]);
}

// x [n,3] -> xpad [n,4] (4th col = 0); one float4 store per row.
__global__ void pad3to4_kernel(const float* __restrict__ in, float* __restrict__ out, int n) {
  int i = blockIdx.x * blockDim.x + threadIdx.x;
  if (i >= n) return;
  float4 v;
  v.x = in[(size_t)i * 3 + 0];
  v.y = in[(size_t)i * 3 + 1];
  v.z = in[(size_t)i * 3 + 2];
  v.w = 0.0f;
  *(float4*)(out + (size_t)i * 4) = v;
}

// W [3,o] -> Wpad [4,o] (row 3 = 0)
__global__ void padW_kernel(const float* __restrict__ in, float* __restrict__ out, int o) {
  int i = blockIdx.x * blockDim.x + threadIdx.x;
  if (i >= 4 * o) return;
  int k = i / o, c = i - k * o;
  out[i] = (k < 3) ? in[(size_t)k * o + c] : 0.0f;
}

// ---------------------------------------------------------------------------
// segment_sum: agg[dst] += feat[src]
// ---------------------------------------------------------------------------
// conv1 flavor: gathers from padded [n,4] rows, scatters into stride-4 agg.
__global__ void segsum_d3p_kernel(const float* __restrict__ xpad,
                                  const int* __restrict__ src,
                                  const int* __restrict__ dst,
                                  float* __restrict__ agg4, int nE) {
  int e = blockIdx.x * blockDim.x + threadIdx.x;
  if (e >= nE) return;
  int s = src[e], t = dst[e];
  const float4 v = *(const float4*)(xpad + (size_t)s * 4);
  float* o = agg4 + (size_t)t * 4;
  atomicAdd(o + 0, v.x);
  atomicAdd(o + 1, v.y);
  atomicAdd(o + 2, v.z);
}

__global__ void segsum_v4_kernel(const float* __restrict__ feat,
                                 const int* __restrict__ src,
                                 const int* __restrict__ dst,
                                 float* __restrict__ agg, int d, int nE) {
  const int dv = d >> 2;                                  // float4 chunks per row
  long long tid = (long long)blockIdx.x * blockDim.x + threadIdx.x;
  if (tid >= (long long)nE * dv) return;
  int e = (int)(tid / dv);
  int c = ((int)(tid - (long long)e * dv)) << 2;
  int s = src[e], t = dst[e];
  const float4 v = *(const float4*)(feat + (size_t)s * d + c);
  float* o = agg + (size_t)t * d + c;
  atomicAdd(o + 0, v.x);
  atomicAdd(o + 1, v.y);
  atomicAdd(o + 2, v.z);
  atomicAdd(o + 3, v.w);
}

// ---------------------------------------------------------------------------
// Inner WMMA tile body shared by both GEMM kernels.
// One wave computes a 16-row x 64-col tile: acc[s] covers cols colBase+16*s.
// ---------------------------------------------------------------------------
template <int D, int O>
__device__ __forceinline__ void tile_accumulate(const float* __restrict__ xp,
                                                const float* __restrict__ gp,
                                                const float* __restrict__ wrp,
                                                const float* __restrict__ wnp,
                                                v8f acc[4]) {
#pragma unroll 4
  for (int kb = 0; kb < D; kb += 4) {
    const v2f ax = *(const v2f*)(xp + kb);          // A frag, X path (b64 load)
    const v2f ag = *(const v2f*)(gp + kb);          // A frag, AGG path
    const float* wr0 = wrp + (size_t)kb * O;
    const float* wn0 = wnp + (size_t)kb * O;
#pragma unroll
    for (int s = 0; s < 4; ++s) {
      v2f br, bn;
      br[0] = wr0[s << 4];
      br[1] = wr0[O + (s << 4)];
      bn[0] = wn0[s << 4];
      bn[1] = wn0[O + (s << 4)];
      acc[s] = wmma4(ax, br, acc[s]);
      acc[s] = wmma4(ag, bn, acc[s]);
    }
  }
}

// H = relu(X @ Wr + AGG @ Wn + b)      (one wave -> 16 rows x 64 cols)
template <int D, int O>
__global__ void gemm_dual_store_kernel(const float* __restrict__ X,
                                       const float* __restrict__ AGG,
                                       const float* __restrict__ Wr,
                                       const float* __restrict__ Wn,
                                       const float* __restrict__ bias,
                                       float* __restrict__ H, int N) {
  const int lane = threadIdx.x & 31;
  const int wid  = (int)((blockIdx.x * blockDim.x + threadIdx.x) >> 5);
  const int rowTiles = N >> 4;
  constexpr int colStrips = O >> 6;
  if (wid >= rowTiles * colStrips) return;
  const int rowTile = wid % rowTiles;
  const int strip   = wid / rowTiles;
  const int lm = lane & 15;
  const int hi = lane >> 4;
  const int row = (rowTile << 4) + lm;
  const int colBase = (strip << 6) + lm;

  const float* xp  = X   + (size_t)row * D + (hi << 1);
  const float* gp  = AGG + (size_t)row * D + (hi << 1);
  const float* wrp = Wr  + (size_t)(hi << 1) * O + colBase;
  const float* wnp = Wn  + (size_t)(hi << 1) * O + colBase;

  v8f acc[4] = {};
  tile_accumulate<D, O>(xp, gp, wrp, wnp, acc);

  const int rbase = (rowTile << 4) + (hi << 3);
#pragma unroll
  for (int s = 0; s < 4; ++s) {
    const int c = colBase + (s << 4);
    const float bv = bias[c];
#pragma unroll
    for (int r = 0; r < 8; ++r)
      H[(size_t)(rbase + r) * O + c] = fmaxf(acc[s][r] + bv, 0.0f);
  }
}

// Fused conv3 + column-max: never materializes the [N, O] activation.
// Each wave sweeps tilesPerWave row-tiles for one 64-col strip, keeps running
// max in registers, then one atomicMax per column.
template <int D, int O>
__global__ void gemm_dual_colmax_kernel(const float* __restrict__ X,
                                        const float* __restrict__ AGG,
                                        const float* __restrict__ Wr,
                                        const float* __restrict__ Wn,
                                        const float* __restrict__ bias,
                                        unsigned* __restrict__ genc,
                                        int N, int relu, int tilesPerWave) {
  const int lane = threadIdx.x & 31;
  const int wid  = (int)((blockIdx.x * blockDim.x + threadIdx.x) >> 5);
  const int rowTiles = N >> 4;
  const int chunks   = rowTiles / tilesPerWave;
  constexpr int colStrips = O >> 6;
  if (wid >= chunks * colStrips) return;
  const int chunk = wid % chunks;
  const int strip = wid / chunks;
  const int lm = lane & 15;
  const int hi = lane >> 4;
  const int colBase = (strip << 6) + lm;

  const float* wrp = Wr + (size_t)(hi << 1) * O + colBase;
  const float* wnp = Wn + (size_t)(hi << 1) * O + colBase;

  float mx[4];
#pragma unroll
  for (int s = 0; s < 4; ++s) mx[s] = -__builtin_inff();

  for (int t = 0; t < tilesPerWave; ++t) {
    const int rowTile = chunk * tilesPerWave + t;
    const size_t rowoff = (size_t)((rowTile << 4) + lm) * D + (hi << 1);
    v8f acc[4] = {};
    tile_accumulate<D, O>(X + rowoff, AGG + rowoff, wrp, wnp, acc);
#pragma unroll
    for (int s = 0; s < 4; ++s)
#pragma unroll
      for (int r = 0; r < 8; ++r) mx[s] = fmaxf(mx[s], acc[s][r]);
  }
#pragma unroll
  for (int s = 0; s < 4; ++s) {
    float v = mx[s] + bias[colBase + (s << 4)];  // per-column bias commutes with max
    if (relu) v = fmaxf(v, 0.0f);                // relu monotonic: relu(max)==max(relu)
    v = fmaxf(v, __shfl_xor(v, 16, 32));         // lanes L and L+16 share the column
    if (lane < 16) atomicMax(&genc[(strip << 6) + (s << 4) + lm], fenc(v));
  }
}

// ---------------------------------------------------------------------------
// out[j] = act(sum_k v[k] * W[k*o + j] + b[j])   (tiny MLP head)
// ---------------------------------------------------------------------------
__global__ void fc_kernel(const float* __restrict__ v, const float* __restrict__ W,
                          const float* __restrict__ b, float* __restrict__ out,
                          int i, int o, int relu) {
  int j = blockIdx.x * blockDim.x + threadIdx.x;
  if (j >= o) return;
  float s = b[j];
  for (int k = 0; k < i; ++k) s = fmaf(v[k], W[(size_t)k * o + j], s);
  out[j] = relu ? fmaxf(s, 0.0f) : s;
}

// d_out[0..1023] = decoded main-path global max; d_out[1024..1032] = fc3 + I3
__global__ void assemble_kernel(const unsigned* __restrict__ gencMain,
                                const float* __restrict__ t9,
                                float* __restrict__ out) {
  int j = blockIdx.x * blockDim.x + threadIdx.x;
  if (j < 1024) {
    out[j] = fdec(gencMain[j]);
  } else if (j < 1033) {
    int t = j - 1024;
    out[j] = t9[t] + ((t == 0 || t == 4 || t == 8) ? 1.0f : 0.0f);
  }
}

// ---------------------------------------------------------------------------
// Host side
// ---------------------------------------------------------------------------
static void launch_conv_stack(const float* xpad, const int* src, const int* dst,
                              int N, int E,
                              const float* c1Wn4, const float* c1Wr4, const float* c1b,
                              const float* c2Wn, const float* c2Wr, const float* c2b,
                              const float* c3Wn, const float* c3Wr, const float* c3b,
                              float* agg, float* h1, float* h2, unsigned* genc,
                              int relu3, hipStream_t stream) {
  const int B = 256;
  // conv1: (padded) 4 -> 64
  zero_f32_kernel<<<(N * 4 + B - 1) / B, B, 0, stream>>>(agg, N * 4);
  segsum_d3p_kernel<<<(E + B - 1) / B, B, 0, stream>>>(xpad, src, dst, agg, E);
  {
    int waves = (N >> 4) * 1;
    gemm_dual_store_kernel<4, 64><<<(waves + 7) / 8, B, 0, stream>>>(
        xpad, agg, c1Wr4, c1Wn4, c1b, h1, N);
  }
  // conv2: 64 -> 128
  zero_f32_kernel<<<(N * 64 + B - 1) / B, B, 0, stream>>>(agg, N * 64);
  {
    long long th = (long long)E * 16;
    segsum_v4_kernel<<<(unsigned)((th + B - 1) / B), B, 0, stream>>>(h1, src, dst, agg, 64, E);
  }
  {
    int waves = (N >> 4) * 2;
    gemm_dual_store_kernel<64, 128><<<(waves + 7) / 8, B, 0, stream>>>(
        h1, agg, c2Wr, c2Wn, c2b, h2, N);
  }
  // conv3: 128 -> 1024, fused with global column-max
  zero_f32_kernel<<<(N * 128 + B - 1) / B, B, 0, stream>>>(agg, N * 128);
  {
    long long th = (long long)E * 32;
    segsum_v4_kernel<<<(unsigned)((th + B - 1) / B), B, 0, stream>>>(h2, src, dst, agg, 128, E);
  }
  zero_u32_kernel<<<(1024 + B - 1) / B, B, 0, stream>>>(genc, 1024);
  {
    int rowTiles = N >> 4;
    int tpw = 25;
    while (tpw > 1 && (rowTiles % tpw)) --tpw;   // N=100000 -> 6250 tiles -> 250 chunks
    int waves = (rowTiles / tpw) * 16;
    gemm_dual_colmax_kernel<128, 1024><<<(waves + 7) / 8, B, 0, stream>>>(
        h2, agg, c3Wr, c3Wn, c3b, genc, N, relu3, tpw);
  }
}

extern "C" void kernel_launch(void* const* d_in, const int* in_sizes, int n_in,
                              void* d_out, int out_size, void* d_ws, size_t ws_size,
                              hipStream_t stream) {
  (void)n_in; (void)out_size; (void)ws_size;
  const float* x  = (const float*)d_in[0];
  const int*   ei = (const int*)d_in[1];
  const int N = in_sizes[0] / 3;
  const int E = in_sizes[1] / 2;
  const int* src = ei;
  const int* dst = ei + E;

  // params flattened alphabetically: conv1..3 (Wn,Wr,b), then stn
  const float *c1Wn=(const float*)d_in[2],  *c1Wr=(const float*)d_in[3],  *c1b=(const float*)d_in[4];
  const float *c2Wn=(const float*)d_in[5],  *c2Wr=(const float*)d_in[6],  *c2b=(const float*)d_in[7];
  const float *c3Wn=(const float*)d_in[8],  *c3Wr=(const float*)d_in[9],  *c3b=(const float*)d_in[10];
  const float *s1Wn=(const float*)d_in[11], *s1Wr=(const float*)d_in[12], *s1b=(const float*)d_in[13];
  const float *s2Wn=(const float*)d_in[14], *s2Wr=(const float*)d_in[15], *s2b=(const float*)d_in[16];
  const float *s3Wn=(const float*)d_in[17], *s3Wr=(const float*)d_in[18], *s3b=(const float*)d_in[19];
  const float *f1W=(const float*)d_in[20],  *f1b=(const float*)d_in[21];
  const float *f2W=(const float*)d_in[22],  *f2b=(const float*)d_in[23];
  const float *f3W=(const float*)d_in[24],  *f3b=(const float*)d_in[25];

  // workspace layout (floats)
  float* agg  = (float*)d_ws;                  // N*128
  float* h1   = agg + (size_t)N * 128;         // N*64
  float* h2   = h1 + (size_t)N * 64;           // N*128
  float* xpad = h2 + (size_t)N * 128;          // N*4
  unsigned* gencS = (unsigned*)(xpad + (size_t)N * 4);  // 1024
  float* gstn = (float*)(gencS + 1024);        // 1024
  float* fo1  = gstn + 1024;                   // 512
  float* fo2  = fo1 + 512;                     // 256
  float* fo3  = fo2 + 256;                     // 16
  unsigned* gencM = (unsigned*)(fo3 + 16);     // 1024
  float* w4   = (float*)(gencM + 1024);        // 4 * 256: s1Wr4|s1Wn4|c1Wr4|c1Wn4
  float* s1Wr4 = w4;
  float* s1Wn4 = w4 + 256;
  float* c1Wr4 = w4 + 512;
  float* c1Wn4 = w4 + 768;

  const int B = 256;
  // One-time (per call) padding: x -> [N,4], conv1 weights -> [4,64]
  pad3to4_kernel<<<(N + B - 1) / B, B, 0, stream>>>(x, xpad, N);
  padW_kernel<<<1, B, 0, stream>>>(s1Wr, s1Wr4, 64);
  padW_kernel<<<1, B, 0, stream>>>(s1Wn, s1Wn4, 64);
  padW_kernel<<<1, B, 0, stream>>>(c1Wr, c1Wr4, 64);
  padW_kernel<<<1, B, 0, stream>>>(c1Wn, c1Wn4, 64);

  // ---- STN branch (relu before max on conv3) ----
  launch_conv_stack(xpad, src, dst, N, E,
                    s1Wn4, s1Wr4, s1b, s2Wn, s2Wr, s2b, s3Wn, s3Wr, s3b,
                    agg, h1, h2, gencS, /*relu3=*/1, stream);
  decode_kernel<<<4, 256, 0, stream>>>(gencS, gstn, 1024);
  fc_kernel<<<2, 256, 0, stream>>>(gstn, f1W, f1b, fo1, 1024, 512, 1);
  fc_kernel<<<1, 256, 0, stream>>>(fo1,  f2W, f2b, fo2, 512,  256, 1);
  fc_kernel<<<1, 256, 0, stream>>>(fo2,  f3W, f3b, fo3, 256,  9,   0);

  // ---- Main branch (no relu on conv3 output) ----
  launch_conv_stack(xpad, src, dst, N, E,
                    c1Wn4, c1Wr4, c1b, c2Wn, c2Wr, c2b, c3Wn, c3Wr, c3b,
                    agg, h1, h2, gencM, /*relu3=*/0, stream);

  assemble_kernel<<<(1033 + 255) / 256, 256, 0, stream>>>(gencM, fo3, (float*)d_out);
}